// TTTLinear_37374805409931
// MI455X (gfx1250) — compile-verified
//
#include <hip/hip_runtime.h>
#include <hip/hip_bf16.h>

// ---------------------------------------------------------------------------
// TTT-Linear for MI455X (gfx1250), full fp32 via V_WMMA_F32_16X16X4_F32,
// with double-buffered GLOBAL_LOAD_ASYNC_TO_LDS_B128 tile staging.
// B=4, S=4096, D=2048, H=16, Dh=128, MB=16.
// ---------------------------------------------------------------------------

typedef __attribute__((ext_vector_type(2))) float v2f;
typedef __attribute__((ext_vector_type(8))) float v8f;

#define NHEAD     16
#define DHEAD     128
#define MBS       16
#define LN_EPS    1e-5f
#define ROPE_BASE 10000.0f

// LDS row paddings (floats).
// GP=36: row stride 144B (16B aligned for b128 async writes), (36r+k)%64
//        distinct over 16 consecutive rows -> conflict-free fragment reads.
// WP=132: 16B-aligned rows (b128 RMW of W update), (4r+k)%64 conflict-free.
// KP=132: 16B-aligned rows for async b128 v-tile writes, (4r+k)%64 distinct.
#define GP  36
#define WP  132
#define KP  132

// ---- CDNA5 async global->LDS copy (ASYNCcnt-tracked DMA), inline asm so it
// ---- works on both ROCm 7.2 and upstream toolchains.
__device__ __forceinline__ void async_copy16(const float* g, const float* l)
{
    const unsigned lds_off = (unsigned)(uintptr_t)l;   // generic->LDS offset
    asm volatile("global_load_async_to_lds_b128 %0, %1, off"
                 :: "v"(lds_off), "v"(g) : "memory");
}
__device__ __forceinline__ void wait_async0()
{
    asm volatile("s_wait_asynccnt 0x0" ::: "memory");
}

// ---------------------------------------------------------------------------
// Generic C = A * Bw^T, A:[M,K] rm, Bw:[N,K] rm, fp32 WMMA 16x16x4.
// OUT_MODE 0: C row-major [M,N].
// OUT_MODE 1: scatter to [B,H,S,Dh]: row=(b*S+s), col=h*Dh+d.
// Workgroup tile 128x128, 8 waves; wave w owns rows [16w,16w+16), all cols.
// K staged 32 at a time, double-buffered via async LDS DMA.
// ---------------------------------------------------------------------------
template <int OUT_MODE>
__global__ __launch_bounds__(256) void gemm_xwt_kernel(
    const float* __restrict__ A, const float* __restrict__ Bw,
    float* __restrict__ C, int M, int N, int K, int Sseq)
{
    extern __shared__ float smem[];            // [2][A:128*GP | B:128*GP]

    const int tid  = threadIdx.x;
    const int wave = tid >> 5;
    const int lane = tid & 31;
    const int l16  = lane & 15;
    const int koff = (lane < 16) ? 0 : 2;      // K sub-offset (f32 A/B layout)
    const int m0   = blockIdx.y * 128;
    const int n0   = blockIdx.x * 128;

    const int ldr = tid >> 3;                  // staging row 0..31 (+32*i)
    const int ldc = (tid & 7) * 4;             // staging col 0,4,..,28

    v8f acc[8];
#pragma unroll
    for (int nt = 0; nt < 8; ++nt)
#pragma unroll
        for (int j = 0; j < 8; ++j) acc[nt][j] = 0.0f;

    // Issue first chunk (buf 0).
    {
        float* As = smem;
        float* Bs = smem + 128 * GP;
#pragma unroll
        for (int i = 0; i < 4; ++i) {
            const int r = ldr + 32 * i;
            async_copy16(A  + (size_t)(m0 + r) * K + ldc, &As[r * GP + ldc]);
            async_copy16(Bw + (size_t)(n0 + r) * K + ldc, &Bs[r * GP + ldc]);
        }
    }
    wait_async0();
    __syncthreads();

    int cur = 0;
    for (int kc = 0; kc < K; kc += 32) {
        // Prefetch next chunk into the other buffer (async DMA).
        if (kc + 32 < K) {
            float* As = smem + (cur ^ 1) * (2 * 128 * GP);
            float* Bs = As + 128 * GP;
#pragma unroll
            for (int i = 0; i < 4; ++i) {
                const int r = ldr + 32 * i;
                async_copy16(A  + (size_t)(m0 + r) * K + kc + 32 + ldc,
                             &As[r * GP + ldc]);
                async_copy16(Bw + (size_t)(n0 + r) * K + kc + 32 + ldc,
                             &Bs[r * GP + ldc]);
            }
        }

        // Compute on current buffer.
        const float* As = smem + cur * (2 * 128 * GP);
        const float* Bs = As + 128 * GP;
        const int ar = wave * 16 + l16;        // A fragment row (M = lane%16)
#pragma unroll
        for (int kk = 0; kk < 8; ++kk) {
            v2f a = *(const v2f*)(&As[ar * GP + kk * 4 + koff]);
#pragma unroll
            for (int nt = 0; nt < 8; ++nt) {
                v2f b = *(const v2f*)(&Bs[(nt * 16 + l16) * GP + kk * 4 + koff]);
                acc[nt] = __builtin_amdgcn_wmma_f32_16x16x4_f32(
                    false, a, false, b, (short)0, acc[nt], false, false);
            }
        }

        wait_async0();     // next buffer resident
        __syncthreads();   // all waves done reading cur / writing nxt
        cur ^= 1;
    }

    // Epilogue with hoisted base pointers -> immediate-offset stores.
    const int mbase = m0 + wave * 16 + ((lane < 16) ? 0 : 8);
    if (OUT_MODE == 0) {
#pragma unroll
        for (int nt = 0; nt < 8; ++nt) {
            float* cp = C + (size_t)mbase * N + n0 + nt * 16 + l16;
#pragma unroll
            for (int r = 0; r < 8; ++r) cp[(size_t)r * N] = acc[nt][r];
        }
    } else {
        const int b  = m0 / Sseq;              // whole 128-row tile in one b
        const int sb = mbase - b * Sseq;       // sequence row of r=0
#pragma unroll
        for (int nt = 0; nt < 8; ++nt) {
            const int col = n0 + nt * 16 + l16;
            const int h = col >> 7, d = col & 127;
            float* cp = C + (((size_t)b * NHEAD + h) * Sseq + sb) * DHEAD + d;
#pragma unroll
            for (int r = 0; r < 8; ++r) cp[r * DHEAD] = acc[nt][r];
        }
    }
}

// ---------------------------------------------------------------------------
// TTT inner scan. One workgroup (256 threads, 8 waves) per (b,h).
// The hidden state is kept TRANSPOSED in LDS: WT[e][d] = W[d][e], so
//  * z-phase B fragment (W[k][e], W[k+1][e]) = WT[e][k..k+1]  -> ds_load_b64
//  * grad update WT[e][d0..d0+7] -= g[0..7] is 8 consecutive floats -> b128
// Per mini-batch n:
//   z  = rope(k_n) @ W           (M=16, N=128, K=128)  -> global out (pre-LN)
//   dz = (z - v_n) * inv_n
//   W -= k_n^T @ dz              (M=128, N=128, K=16)
// v tile fetched with async LDS DMA; RoPE'd k built through VALU.
// ---------------------------------------------------------------------------
__global__ __launch_bounds__(256) void ttt_scan_kernel(
    const float* __restrict__ Kg, const float* __restrict__ Vg,
    float* __restrict__ Z, int Sseq)
{
    extern __shared__ float smem[];
    float* WT    = smem;                       // 128 x WP (transposed W)
    float* kbuf  = WT   + 128 * WP;            // 16 x KP  (rope'd keys)
    float* vbuf  = kbuf + MBS * KP;            // 16 x KP
    float* dzbuf = vbuf + MBS * KP;            // 16 x KP

    const int bh   = blockIdx.x;               // 0..B*H-1
    const int b    = bh / NHEAD;
    const int h    = bh % NHEAD;
    const float* kg = Kg + (size_t)bh * Sseq * DHEAD;
    const float* vg = Vg + (size_t)bh * Sseq * DHEAD;

    const int tid  = threadIdx.x;
    const int wave = tid >> 5;
    const int lane = tid & 31;
    const int l16  = lane & 15;
    const int koff = (lane < 16) ? 0 : 2;
    const int Dm   = NHEAD * DHEAD;

    // WT = I (identity is symmetric)
    for (int i = tid; i < 128 * 128; i += 256) {
        const int r = i >> 7, c = i & 127;
        WT[r * WP + c] = (r == c) ? 1.0f : 0.0f;
    }

    const float inv_n = 2.0f / (float)(MBS * DHEAD);
    const float log_base = __logf(ROPE_BASE);
    const int nmb = Sseq / MBS;

    // Per-thread RoPE constants (channel-dependent only) hoisted out of the
    // scan loop: thread handles elements i = tid + 256*j of the 16x128 tile.
    int   jd[8], js[8];
    float jfreq[8], jsign[8];
#pragma unroll
    for (int j = 0; j < 8; ++j) {
        const int i = tid + 256 * j;
        js[j] = i >> 7;
        jd[j] = i & 127;
        const int fi = jd[j] & 63;
        jfreq[j] = __expf(-(float)(2 * fi) * (1.0f / DHEAD) * log_base);
        jsign[j] = (jd[j] < 64) ? -1.0f : 1.0f;
    }

    // v-tile async staging pattern: 512 float4s, 2 per thread.
    const int vs0 = (tid + 0)   >> 5, vc0 = ((tid + 0)   & 31) * 4;
    const int vs1 = (tid + 256) >> 5, vc1 = ((tid + 256) & 31) * 4;

    for (int n = 0; n < nmb; ++n) {
        __syncthreads();   // prev grad done: W updated, k/v/dz bufs free

        // v tile: async DMA straight into LDS.
        async_copy16(vg + (size_t)(n * MBS + vs0) * DHEAD + vc0,
                     &vbuf[vs0 * KP + vc0]);
        async_copy16(vg + (size_t)(n * MBS + vs1) * DHEAD + vc1,
                     &vbuf[vs1 * KP + vc1]);

        // k tile with RoPE (partner element at d^64).
        const float* kgn = kg + (size_t)(n * MBS) * DHEAD;
#pragma unroll
        for (int j = 0; j < 8; ++j) {
            const int s = js[j], d = jd[j];
            const float kr = kgn[s * DHEAD + d];
            const float kp = kgn[s * DHEAD + (d ^ 64)];
            const float ang = (float)(n * MBS + s) * jfreq[j];
            float sn, cs;
            sincosf(ang, &sn, &cs);
            kbuf[s * KP + d] = kr * cs + jsign[j] * kp * sn;
        }
        wait_async0();
        __syncthreads();

        // ---- z phase: wave w owns N(e)-tile [16w, 16w+16) ----
        const int e0 = wave * 16;
        v8f zacc;
#pragma unroll
        for (int j = 0; j < 8; ++j) zacc[j] = 0.0f;
#pragma unroll 4
        for (int kk = 0; kk < 32; ++kk) {
            v2f a = *(const v2f*)(&kbuf[l16 * KP + kk * 4 + koff]);
            v2f w0 = *(const v2f*)(&WT[(e0 + l16) * WP + kk * 4 + koff]);
            zacc = __builtin_amdgcn_wmma_f32_16x16x4_f32(
                false, a, false, w0, (short)0, zacc, false, false);
        }
        {
            const int mbase = (lane < 16) ? 0 : 8;
            float*       zp  = Z + ((size_t)b * Sseq + n * MBS + mbase) * Dm
                                 + h * DHEAD + e0 + l16;
            float*       dzp = &dzbuf[mbase * KP + e0 + l16];
            const float* vp  = &vbuf [mbase * KP + e0 + l16];
#pragma unroll
            for (int r = 0; r < 8; ++r) {
                const float zval = zacc[r];
                zp [r * Dm] = zval;
                dzp[r * KP] = (zval - vp[r * KP]) * inv_n;
            }
        }
        __syncthreads();   // dz complete; all W reads complete

        // ---- grad phase: wave w owns d-stripe [16w, 16w+16) ----
        const int d0 = wave * 16;
        const int mbase = (lane < 16) ? 0 : 8;
#pragma unroll
        for (int et = 0; et < 8; ++et) {
            const int ec0 = et * 16;
            v8f g;
#pragma unroll
            for (int j = 0; j < 8; ++j) g[j] = 0.0f;
#pragma unroll
            for (int kk = 0; kk < 4; ++kk) {
                v2f a;   // A = k^T: M=d, K=s
                a.x = kbuf[(kk * 4 + koff)     * KP + d0 + l16];
                a.y = kbuf[(kk * 4 + koff + 1) * KP + d0 + l16];
                v2f bz;  // B = dz: K=s, N=e
                bz.x = dzbuf[(kk * 4 + koff)     * KP + ec0 + l16];
                bz.y = dzbuf[(kk * 4 + koff + 1) * KP + ec0 + l16];
                g = __builtin_amdgcn_wmma_f32_16x16x4_f32(
                    false, a, false, bz, (short)0, g, false, false);
            }
            // Lane holds N=e fixed, M=d contiguous -> 8 consecutive floats.
            float* wp = &WT[(ec0 + l16) * WP + d0 + mbase];
#pragma unroll
            for (int r = 0; r < 8; ++r) wp[r] -= g[r];   // LR = 1
        }
    }
}

// ---------------------------------------------------------------------------
// In-place LayerNorm over last dim (D=2048), one workgroup per row.
// ---------------------------------------------------------------------------
__global__ __launch_bounds__(256) void ln_kernel(
    float* __restrict__ Y, const float* __restrict__ gamma,
    const float* __restrict__ beta, int D)
{
    const int row = blockIdx.x;
    float* y = Y + (size_t)row * D;
    const int tid = threadIdx.x;

    float v[8];
    float s = 0.0f, s2 = 0.0f;
#pragma unroll
    for (int i = 0; i < 8; ++i) {
        const float x = y[tid + 256 * i];
        v[i] = x; s += x; s2 += x * x;
    }
#pragma unroll
    for (int off = 16; off > 0; off >>= 1) {
        s  += __shfl_down(s,  off, 32);
        s2 += __shfl_down(s2, off, 32);
    }
    __shared__ float ws[8], ws2[8];
    const int wave = tid >> 5, lane = tid & 31;
    if (lane == 0) { ws[wave] = s; ws2[wave] = s2; }
    __syncthreads();
    float ts = 0.0f, ts2 = 0.0f;
#pragma unroll
    for (int w = 0; w < 8; ++w) { ts += ws[w]; ts2 += ws2[w]; }
    const float mean = ts / (float)D;
    const float var  = ts2 / (float)D - mean * mean;
    const float inv  = rsqrtf(var + LN_EPS);
#pragma unroll
    for (int i = 0; i < 8; ++i) {
        const int c = tid + 256 * i;
        y[c] = (v[i] - mean) * inv * gamma[c] + beta[c];
    }
}

// ---------------------------------------------------------------------------
extern "C" void kernel_launch(void* const* d_in, const int* in_sizes, int n_in,
                              void* d_out, int out_size, void* d_ws, size_t ws_size,
                              hipStream_t stream)
{
    const float* x     = (const float*)d_in[0];
    const float* Wk    = (const float*)d_in[1];
    const float* Wv    = (const float*)d_in[2];
    const float* Wo    = (const float*)d_in[3];
    const float* gamma = (const float*)d_in[4];
    const float* beta  = (const float*)d_in[5];
    float* out = (float*)d_out;

    const int Bn = 4, Sseq = 4096, D = 2048;
    const int M = Bn * Sseq;

    float* kws = (float*)d_ws;                  // [B,H,S,Dh]
    float* vws = kws + (size_t)M * D;           // [B,H,S,Dh]
    float* zws = vws + (size_t)M * D;           // [B,S,D] pre-LN / post-LN

    const size_t gemm_lds = (size_t)(2 * 2 * 128 * GP) * sizeof(float);
    const size_t scan_lds = (size_t)(128 * WP + 3 * MBS * KP) * sizeof(float);
    (void)hipFuncSetAttribute((const void*)gemm_xwt_kernel<0>,
                              hipFuncAttributeMaxDynamicSharedMemorySize,
                              (int)gemm_lds);
    (void)hipFuncSetAttribute((const void*)gemm_xwt_kernel<1>,
                              hipFuncAttributeMaxDynamicSharedMemorySize,
                              (int)gemm_lds);
    (void)hipFuncSetAttribute((const void*)ttt_scan_kernel,
                              hipFuncAttributeMaxDynamicSharedMemorySize,
                              (int)scan_lds);

    dim3 gg(D / 128, M / 128), bb(256);
    // 1-2) K/V projections, scattered to per-(b,h) scan layout.
    gemm_xwt_kernel<1><<<gg, bb, gemm_lds, stream>>>(x, Wk, kws, M, D, D, Sseq);
    gemm_xwt_kernel<1><<<gg, bb, gemm_lds, stream>>>(x, Wv, vws, M, D, D, Sseq);
    // 3) TTT scan (RoPE fused into k load), z written in [B,S,D].
    ttt_scan_kernel<<<dim3(Bn * NHEAD), bb, scan_lds, stream>>>(kws, vws, zws, Sseq);
    // 4) In-place LayerNorm.
    ln_kernel<<<dim3(M), bb, 0, stream>>>(zws, gamma, beta, D);
    // 5) Output projection.
    gemm_xwt_kernel<0><<<gg, bb, gemm_lds, stream>>>(zws, Wo, out, M, D, D, Sseq);
}